// PRUNNRT_8229157339178
// MI455X (gfx1250) — compile-verified
//
#include <hip/hip_runtime.h>
#include <hip/hip_bf16.h>

typedef float v2f __attribute__((ext_vector_type(2)));
typedef float v8f __attribute__((ext_vector_type(8)));

#define NF     9
#define NBATCH 64
#define TEMP   0.1f
#define FEPS   1e-5f

// One workgroup does everything: BN stats -> 2-way softmax -> sparse leaf
// gather (512 of 5^9 leaves) -> build P^T (512x64 fp32 weights) in LDS ->
// 64x512 @ 512x1 GEMM on v_wmma_f32_16x16x4_f32 -> sigmoid.
__global__ __launch_bounds__(512)
void soft_tree_kernel(const float* __restrict__ x,
                      const float* __restrict__ cut,
                      const float* __restrict__ bnw,
                      const float* __restrict__ bnb,
                      const float* __restrict__ leaf,
                      float* __restrict__ out)
{
    __shared__ float xs[NF][NBATCH];          // x transposed
    __shared__ float scaleF[NF], shiftF[NF], cF[NF];
    __shared__ float pL[NBATCH][NF][2];       // softmax probs
    __shared__ float Lg[512];                 // gathered leaf scores
    __shared__ float Pt[512][NBATCH + 1];     // P^T, padded: stride 65 words
    __shared__ float partial[4][NBATCH];      // K-slice partial sums

    const int tid  = threadIdx.x;
    const int lane = tid & 31;
    const int wave = tid >> 5;

    // ---- Phase 1: load x (row-major (64,9)) transposed into LDS ----
    for (int t = tid; t < NBATCH * NF; t += 512) {
        int b = t / NF;
        int f = t - b * NF;
        xs[f][b] = x[t];
    }
    __syncthreads();

    // ---- Phase 2: batchnorm stats per feature (population var) ----
    if (tid < NF) {
        float s = 0.f, s2 = 0.f;
        for (int b = 0; b < NBATCH; ++b) {
            float v = xs[tid][b];
            s += v; s2 += v * v;
        }
        float mean = s * (1.0f / NBATCH);
        float var  = s2 * (1.0f / NBATCH) - mean * mean;
        float sc   = bnw[tid] * rsqrtf(var + FEPS);
        scaleF[tid] = sc;
        shiftF[tid] = bnb[tid] - mean * sc;
        cF[tid]     = cut[tid * 4];           // cut_points[:, 0]
    }
    __syncthreads();

    // ---- Phase 3: 2-way softmax per (batch, feature) ----
    for (int t = tid; t < NBATCH * NF; t += 512) {
        int b = t / NF;
        int f = t - b * NF;
        float xn = xs[f][b] * scaleF[f] + shiftF[f];
        float h0 = xn * (1.0f / TEMP);
        float h1 = (2.0f * xn - cF[f]) * (1.0f / TEMP);
        float mx = fmaxf(h0, h1);
        float e0 = __expf(h0 - mx);
        float e1 = __expf(h1 - mx);
        float inv = 1.0f / (e0 + e1);
        pL[b][f][0] = e0 * inv;
        pL[b][f][1] = e1 * inv;
    }
    // ---- Phase 3b: gather the 512 reachable leaf scores ----
    // combo m: feature i <-> bit (8-i); flat idx = sum bit * 5^(8-i)
    if (tid < 512) {
        int idx = 0;
        #pragma unroll
        for (int i = 0; i < NF; ++i)
            idx = idx * 5 + ((tid >> (8 - i)) & 1);
        Lg[tid] = leaf[idx];
    }
    __syncthreads();

    // ---- Phase 4: build P^T[m][b]; wave handles 4 batches ----
    // m = j*32 + lane : lane bits 4..0 -> features 4..8 ("base"),
    //                   j    bits 3..0 -> features 0..3 ("head")
    for (int bb = 0; bb < 4; ++bb) {
        int b = wave * 4 + bb;
        float p0[NF], p1[NF];
        #pragma unroll
        for (int f = 0; f < NF; ++f) { p0[f] = pL[b][f][0]; p1[f] = pL[b][f][1]; }

        float base = 1.0f;
        #pragma unroll
        for (int f = 4; f < 9; ++f)
            base *= ((lane >> (8 - f)) & 1) ? p1[f] : p0[f];

        float h[16];
        h[0] = 1.0f;
        int sz = 1;
        #pragma unroll
        for (int f = 0; f < 4; ++f) {
            #pragma unroll
            for (int k2 = 15; k2 >= 0; --k2) {
                if (k2 < sz) {
                    float a = h[k2];
                    h[2 * k2 + 0] = a * p0[f];
                    h[2 * k2 + 1] = a * p1[f];
                }
            }
            sz *= 2;
        }
        #pragma unroll
        for (int j = 0; j < 16; ++j)
            Pt[j * 32 + lane][b] = h[j] * base;   // stride-65: conflict-free
    }
    __syncthreads();

    // ---- Phase 5: GEMM out = P(64x512) @ Lg(512x1) via WMMA f32 16x16x4 ----
    // 16 waves = 4 batch-tiles x 4 K-slices of 128.
    {
        const int tile = wave & 3;
        const int ks   = wave >> 2;
        const int k0   = ks * 128;
        const int row  = tile * 16 + (lane & 15);
        const int c0   = (lane < 16) ? 0 : 2;     // K-half per A/B layout

        v8f acc = {0.f, 0.f, 0.f, 0.f, 0.f, 0.f, 0.f, 0.f};
        for (int k = k0; k < k0 + 128; k += 4) {
            v2f a, bv;
            a.x = Pt[k + c0][row];
            a.y = Pt[k + c0 + 1][row];
            // Broadcast-load the B pair into ALL lanes (one ds_load_b64,
            // same address per half-wave). Lanes N!=0 replicate column 0,
            // which only affects D columns we never read.
            bv = *(const v2f*)&Lg[k + c0];
            acc = __builtin_amdgcn_wmma_f32_16x16x4_f32(
                false, a, false, bv, (short)0, acc, false, false);
        }
        // D column 0: lane 0 holds M=0..7 in acc[0..7], lane 16 holds M=8..15
        if ((lane & 15) == 0) {
            int half = lane >> 4;
            #pragma unroll
            for (int j = 0; j < 8; ++j)
                partial[ks][tile * 16 + half * 8 + j] = acc[j];
        }
    }
    __syncthreads();

    // ---- Phase 6: reduce K-slices + sigmoid ----
    if (tid < NBATCH) {
        float s = partial[0][tid] + partial[1][tid]
                + partial[2][tid] + partial[3][tid];
        out[tid] = 1.0f / (1.0f + __expf(-s));
    }
}

extern "C" void kernel_launch(void* const* d_in, const int* in_sizes, int n_in,
                              void* d_out, int out_size, void* d_ws, size_t ws_size,
                              hipStream_t stream) {
    const float* x    = (const float*)d_in[0];
    const float* cut  = (const float*)d_in[1];
    const float* bnw  = (const float*)d_in[2];
    const float* bnb  = (const float*)d_in[3];
    const float* leaf = (const float*)d_in[4];
    float* out = (float*)d_out;
    (void)in_sizes; (void)n_in; (void)out_size; (void)d_ws; (void)ws_size;
    soft_tree_kernel<<<dim3(1), dim3(512), 0, stream>>>(x, cut, bnw, bnb, leaf, out);
}